// Simulation_20495583936963
// MI455X (gfx1250) — compile-verified
//
#include <hip/hip_runtime.h>
#include <math.h>

// ---------------------------------------------------------------------------
// Cable-through-tet-mesh simulation, MI455X (gfx1250).
//
// Structure: forces only touch <=512 node slots (128 holes x 4 bary nodes).
//   Kernel 1 (sim_bulk):  all 20000 nodes, ballistic 500-step recurrence in
//                         registers (exact sequential rounding order).
//   Kernel 2 (sim_holes): one workgroup (4 wave32) keeps the 512-slot coupled
//                         subsystem in LDS for all 500 steps, then overwrites
//                         the forced nodes' final state. Deterministic
//                         (duplicate-index chains instead of float atomics).
// CDNA5-specific paths: global_load_async_to_lds_b128 + s_wait_asynccnt
// (ASYNCcnt, ISA ch.10/15.18), wave32 workgroup barriers, LDS ds ops.
// ---------------------------------------------------------------------------

namespace {
constexpr int   kNodes = 20000;
constexpr int   kHoles = 128;
constexpr int   kSlots = kHoles * 4;   // 512
constexpr int   kSteps = 500;
constexpr float kDt    = 1e-3f;
constexpr float kStiff = 100.0f;
constexpr float kDamp  = 1.0f;
constexpr float kGz    = -9.81f;
}

// ---------------------------------------------------------------------------
// Kernel 1: ballistic update for every node (forces == 0 contribution).
// Reference per step: nv = v + DT*(0*im + g); np = p + DT*nv.
// x/y velocity adds DT*0 == +0.0 each step (no change); replay z and the
// position accumulation sequentially to match reference rounding.
// ---------------------------------------------------------------------------
__global__ __launch_bounds__(256) void sim_bulk(const float* __restrict__ p0,
                                                const float* __restrict__ v0,
                                                float* __restrict__ out)
{
  int i = blockIdx.x * blockDim.x + threadIdx.x;
  if (i >= kNodes) return;

  float px = p0[3*i+0], py = p0[3*i+1], pz = p0[3*i+2];
  float vx = v0[3*i+0], vy = v0[3*i+1], vz = v0[3*i+2];
  const float dtgz = __fmul_rn(kDt, kGz);

  #pragma unroll 5
  for (int s = 0; s < kSteps; ++s) {
    vz = __fadd_rn(vz, dtgz);                      // v += DT*g   (z only)
    px = __fadd_rn(px, __fmul_rn(kDt, vx));        // p += DT*v
    py = __fadd_rn(py, __fmul_rn(kDt, vy));
    pz = __fadd_rn(pz, __fmul_rn(kDt, vz));
  }

  out[3*i+0] = px; out[3*i+1] = py; out[3*i+2] = pz;
  float* ov = out + 3*kNodes;
  ov[3*i+0] = vx; ov[3*i+1] = vy; ov[3*i+2] = vz;
}

// ---------------------------------------------------------------------------
// Kernel 2: coupled 512-slot subsystem, 1 block x 128 threads (thread h owns
// hole h and slots 4h..4h+3). State lives in LDS for the whole 500-step loop.
// ---------------------------------------------------------------------------
__global__ __launch_bounds__(128) void sim_holes(const float* __restrict__ node_pos,
                                                 const float* __restrict__ node_vel,
                                                 const int*   __restrict__ hole_idx,
                                                 const float* __restrict__ hole_w,
                                                 const float* __restrict__ inv_mass,
                                                 float* __restrict__ out)
{
  __shared__ __align__(16) int   idx_sh[kSlots];
  __shared__ __align__(16) float w_sh[kSlots];
  __shared__ short canon_sh[kSlots];   // first slot with same node index
  __shared__ short next_sh[kSlots];    // next duplicate slot (-1 = none)
  __shared__ float px_sh[kSlots], py_sh[kSlots], pz_sh[kSlots];
  __shared__ float vx_sh[kSlots], vy_sh[kSlots], vz_sh[kSlots];
  __shared__ float fx_sh[kSlots], fy_sh[kSlots], fz_sh[kSlots];
  __shared__ float im_sh[kSlots];
  __shared__ float hpx[kHoles], hpy[kHoles], hpz[kHoles];

  const int h = threadIdx.x;           // 0..127

  // --- Stage hole_idx / hole_w into LDS via CDNA5 async-to-LDS DMA path ---
  {
    uint32_t lds_i = (uint32_t)(uintptr_t)(&idx_sh[4*h]);
    uint32_t lds_w = (uint32_t)(uintptr_t)(&w_sh[4*h]);
    uint64_t ga_i  = (uint64_t)(uintptr_t)(hole_idx + 4*h);   // 16B aligned
    uint64_t ga_w  = (uint64_t)(uintptr_t)(hole_w   + 4*h);
    asm volatile("global_load_async_to_lds_b128 %0, %1, off"
                 :: "v"(lds_i), "v"(ga_i) : "memory");
    asm volatile("global_load_async_to_lds_b128 %0, %1, off"
                 :: "v"(lds_w), "v"(ga_w) : "memory");
    asm volatile("s_wait_asynccnt 0" ::: "memory");
  }
  __syncthreads();

  // --- Init: load slot states, build canonical/duplicate chains (once) ---
  for (int k = 0; k < 4; ++k) {
    int j = 4*h + k;
    int n = idx_sh[j];
    px_sh[j] = node_pos[3*n+0]; py_sh[j] = node_pos[3*n+1]; pz_sh[j] = node_pos[3*n+2];
    vx_sh[j] = node_vel[3*n+0]; vy_sh[j] = node_vel[3*n+1]; vz_sh[j] = node_vel[3*n+2];
    im_sh[j] = inv_mass[n];

    int c = j;
    for (int q = 0; q < j; ++q)      { if (idx_sh[q] == n) { c = q; break; } }
    canon_sh[j] = (short)c;
    int nx = -1;
    for (int q = j+1; q < kSlots; ++q){ if (idx_sh[q] == n) { nx = q; break; } }
    next_sh[j] = (short)nx;
  }
  __syncthreads();

  // --- 500-step loop, fully LDS-resident ---
  for (int step = 0; step < kSteps; ++step) {
    // Phase A: barycentric gather of this hole's position/velocity.
    float hx = 0.f, hy = 0.f, hz = 0.f;        // hole position
    float ux = 0.f, uy = 0.f, uz = 0.f;        // hole velocity
    for (int k = 0; k < 4; ++k) {
      int j = 4*h + k; int c = canon_sh[j]; float w = w_sh[j];
      hx += w * px_sh[c]; hy += w * py_sh[c]; hz += w * pz_sh[c];
      ux += w * vx_sh[c]; uy += w * vy_sh[c]; uz += w * vz_sh[c];
    }
    hpx[h] = hx; hpy[h] = hy; hpz[h] = hz;
    __syncthreads();

    // Phase B: cable tension (unit-vector along segments) + damping.
    float fx = 0.f, fy = 0.f, fz = 0.f;
    if (h < kHoles - 1) {                       // + STIFF * d_h
      float sx = hpx[h+1] - hx, sy = hpy[h+1] - hy, sz = hpz[h+1] - hz;
      float dn = sqrtf(sx*sx + sy*sy + sz*sz) + 1e-8f;
      fx += kStiff * (sx / dn);
      fy += kStiff * (sy / dn);
      fz += kStiff * (sz / dn);
    }
    if (h > 0) {                                // - STIFF * d_{h-1}
      float sx = hx - hpx[h-1], sy = hy - hpy[h-1], sz = hz - hpz[h-1];
      float dn = sqrtf(sx*sx + sy*sy + sz*sz) + 1e-8f;
      fx -= kStiff * (sx / dn);
      fy -= kStiff * (sy / dn);
      fz -= kStiff * (sz / dn);
    }
    fx -= kDamp * ux; fy -= kDamp * uy; fz -= kDamp * uz;

    for (int k = 0; k < 4; ++k) {               // per-slot scatter (no atomics)
      int j = 4*h + k; float w = w_sh[j];
      fx_sh[j] = w * fx; fy_sh[j] = w * fy; fz_sh[j] = w * fz;
    }
    __syncthreads();

    // Phase C: integrate canonical slots (sum duplicate chain for exact B^T f).
    for (int k = 0; k < 4; ++k) {
      int j = 4*h + k;
      if ((int)canon_sh[j] == j) {
        float ax = 0.f, ay = 0.f, az = 0.f;
        int q = j;
        while (q >= 0) { ax += fx_sh[q]; ay += fy_sh[q]; az += fz_sh[q]; q = next_sh[q]; }
        float im = im_sh[j];
        float nvx = __fadd_rn(vx_sh[j], __fmul_rn(kDt, __fadd_rn(__fmul_rn(ax, im), 0.0f)));
        float nvy = __fadd_rn(vy_sh[j], __fmul_rn(kDt, __fadd_rn(__fmul_rn(ay, im), 0.0f)));
        float nvz = __fadd_rn(vz_sh[j], __fmul_rn(kDt, __fadd_rn(__fmul_rn(az, im), kGz)));
        vx_sh[j] = nvx; vy_sh[j] = nvy; vz_sh[j] = nvz;
        px_sh[j] = __fadd_rn(px_sh[j], __fmul_rn(kDt, nvx));
        py_sh[j] = __fadd_rn(py_sh[j], __fmul_rn(kDt, nvy));
        pz_sh[j] = __fadd_rn(pz_sh[j], __fmul_rn(kDt, nvz));
      }
    }
    __syncthreads();
  }

  // --- Overwrite forced nodes' final state (canonical slots only: one
  //     writer per unique node index) ---
  float* ov = out + 3*kNodes;
  for (int k = 0; k < 4; ++k) {
    int j = 4*h + k;
    if ((int)canon_sh[j] == j) {
      int n = idx_sh[j];
      out[3*n+0] = px_sh[j]; out[3*n+1] = py_sh[j]; out[3*n+2] = pz_sh[j];
      ov [3*n+0] = vx_sh[j]; ov [3*n+1] = vy_sh[j]; ov [3*n+2] = vz_sh[j];
    }
  }
}

// ---------------------------------------------------------------------------
extern "C" void kernel_launch(void* const* d_in, const int* in_sizes, int n_in,
                              void* d_out, int out_size, void* d_ws, size_t ws_size,
                              hipStream_t stream)
{
  (void)in_sizes; (void)n_in; (void)out_size; (void)d_ws; (void)ws_size;
  const float* node_pos = (const float*)d_in[0];
  const float* node_vel = (const float*)d_in[1];
  const int*   hole_idx = (const int*)  d_in[2];
  const float* hole_w   = (const float*)d_in[3];
  const float* inv_mass = (const float*)d_in[4];
  float* out = (float*)d_out;

  sim_bulk <<<(kNodes + 255) / 256, 256, 0, stream>>>(node_pos, node_vel, out);
  sim_holes<<<1, kHoles, 0, stream>>>(node_pos, node_vel, hole_idx, hole_w,
                                      inv_mass, out);
}